// lveg_1108101562416
// MI455X (gfx1250) — compile-verified
//
#include <hip/hip_runtime.h>
#include <math.h>

// ---------------------------------------------------------------------------
// Problem constants (from reference: B=2, T=128, L=46, G=2)
// ---------------------------------------------------------------------------
static constexpr int B   = 2;
static constexpr int T   = 128;
static constexpr int L   = 46;
static constexpr int G   = 2;
static constexpr int BT  = B * T;          // 256
static constexpr int LG  = L * G;          // 92
static constexpr int LL  = L * L;          // 2116
static constexpr int LLG = L * L * G;      // 4232
static constexpr int LLL = L * L * L;      // 97336

// workspace layout (floats):
//  [0,      LLG)  tpm  = tanh(trans_p_mu)                 [c][q][g]
//  [LLG,   2LLG)  e2   = exp(2*tanh(trans_p_var))         [c][q][g]
//  [2LLG,  3LLG)  tcm  = tanh(trans_c_mu)                 [p][c][g]
//  [3LLG,  4LLG)  etc  = exp(2*tcm)                       [p][c][g]
//  [4LLG,  4LLG+BT*LG)        sm  = tanh(s_mu gather)     [bt][c][g]
//  [4LLG+BT*LG, 4LLG+2BT*LG)  es  = exp(2*tanh(s_var))    [bt][c][g]
// total 4*4232 + 2*23552 = 64032 floats = 256128 bytes of d_ws
static constexpr int WS_TPM = 0;
static constexpr int WS_E2  = LLG;
static constexpr int WS_TCM = 2 * LLG;
static constexpr int WS_ETC = 3 * LLG;
static constexpr int WS_SM  = 4 * LLG;
static constexpr int WS_ES  = 4 * LLG + BT * LG;

#ifndef __has_builtin
#define __has_builtin(x) 0
#endif

#if __has_builtin(__builtin_amdgcn_global_load_async_to_lds_b32)
#define HAVE_ASYNC_LDS 1
#else
#define HAVE_ASYNC_LDS 0
#endif
#if __has_builtin(__builtin_amdgcn_global_load_async_to_lds_b128)
#define HAVE_ASYNC_LDS_B128 1
#else
#define HAVE_ASYNC_LDS_B128 0
#endif
#if __has_builtin(__builtin_amdgcn_s_wait_asynccnt)
#define HAVE_WAIT_ASYNC 1
#else
#define HAVE_WAIT_ASYNC 0
#endif

#define AS1 __attribute__((address_space(1)))
#define AS3 __attribute__((address_space(3)))

typedef int   vint4   __attribute__((ext_vector_type(4)));
typedef float vfloat4 __attribute__((ext_vector_type(4)));

typedef AS1 int*   g_as_ptr;    // global (addrspace 1) scalar pointer (b32 form)
typedef AS3 int*   l_as_ptr;    // LDS (addrspace 3) scalar pointer
typedef AS1 vint4* g4_as_ptr;   // global vector pointer (b128 form)
typedef AS3 vint4* l4_as_ptr;   // LDS vector pointer

// -------------------------- fast transcendentals ---------------------------
static __device__ __forceinline__ float fast_tanh(float x) {
#if __has_builtin(__builtin_amdgcn_tanhf)
    return __builtin_amdgcn_tanhf(x);            // v_tanh_f32 (gfx1250)
#else
    return tanhf(x);
#endif
}
static __device__ __forceinline__ float fast_exp(float x) {  // e^x
#if __has_builtin(__builtin_amdgcn_exp2f)
    return __builtin_amdgcn_exp2f(x * 1.4426950408889634f);  // v_exp_f32
#else
    return exp2f(x * 1.4426950408889634f);
#endif
}
static __device__ __forceinline__ float fast_log(float x) {  // ln(x)
#if __has_builtin(__builtin_amdgcn_logf)
    return __builtin_amdgcn_logf(x) * 0.6931471805599453f;   // v_log_f32
#else
    return log2f(x) * 0.6931471805599453f;
#endif
}
static __device__ __forceinline__ float fast_rcp(float x) {
#if __has_builtin(__builtin_amdgcn_rcpf)
    return __builtin_amdgcn_rcpf(x);             // v_rcp_f32
#else
    return 1.0f / x;
#endif
}

// ------------------------- async global -> LDS copy ------------------------
// 16-byte transfer; all callers guarantee 16B alignment of src and dst.
static __device__ __forceinline__ void copy_g2lds_v4(const float* g, float* l) {
#if HAVE_ASYNC_LDS_B128
    __builtin_amdgcn_global_load_async_to_lds_b128(
        (g4_as_ptr)(g), (l4_as_ptr)(l), 0, 0);   // global_load_async_to_lds_b128
#elif HAVE_ASYNC_LDS
#pragma unroll
    for (int e = 0; e < 4; ++e)
        __builtin_amdgcn_global_load_async_to_lds_b32(
            (g_as_ptr)(g + e), (l_as_ptr)(l + e), 0, 0);
#else
    *(vfloat4*)l = *(const vfloat4*)g;
#endif
}
static __device__ __forceinline__ void async_wait0() {
#if HAVE_WAIT_ASYNC
    __builtin_amdgcn_s_wait_asynccnt(0);
#elif HAVE_ASYNC_LDS || HAVE_ASYNC_LDS_B128
    asm volatile("s_wait_asynccnt 0" ::: "memory");
#endif
}

// ---------------------------------------------------------------------------
// Prep 1: tanh/exp of transition tables (4232 elements)
// ---------------------------------------------------------------------------
__global__ void lveg_prep_trans(const float* __restrict__ tpmu,
                                const float* __restrict__ tpvar,
                                const float* __restrict__ tcmu,
                                float* __restrict__ wsf) {
    int i = blockIdx.x * blockDim.x + threadIdx.x;
    if (i >= LLG) return;
    float tpm = fast_tanh(tpmu[i]);
    float tpv = fast_tanh(tpvar[i]);
    float tcm = fast_tanh(tcmu[i]);      // note: t_c_var == t_c_mu in reference
    wsf[WS_TPM + i] = tpm;
    wsf[WS_E2  + i] = fast_exp(2.0f * tpv);
    wsf[WS_TCM + i] = tcm;
    wsf[WS_ETC + i] = fast_exp(2.0f * tcm);
}

// ---------------------------------------------------------------------------
// Prep 2: embedding gather + tanh (23552 elements)
// ---------------------------------------------------------------------------
__global__ void lveg_prep_emb(const int* __restrict__ input,
                              const float* __restrict__ smt,
                              const float* __restrict__ svt,
                              float* __restrict__ wsf) {
    int i = blockIdx.x * blockDim.x + threadIdx.x;   // over BT*LG
    if (i >= BT * LG) return;
    int bt = i / LG;
    int j  = i - bt * LG;
    long long row = (long long)input[bt] * LG;
    wsf[WS_SM + i] = fast_tanh(smt[row + j]);
    wsf[WS_ES + i] = fast_exp(2.0f * fast_tanh(svt[row + j]));
}

// ---------------------------------------------------------------------------
// Main: one block per (p, chunk of 8 timesteps). Async-stages t_p tables
// (34 KB) into LDS with b128 transfers, builds the per-(b,t) child-state
// product for this parent p, then streams 529 float4 NT stores per (b,t).
// ---------------------------------------------------------------------------
static constexpr int TCHUNK = 8;
static constexpr int NCHUNK = BT / TCHUNK;   // 32
static constexpr int NTHR   = 256;

__global__ __launch_bounds__(NTHR) void lveg_main(const float* __restrict__ wsf,
                                                  const float* __restrict__ mask,
                                                  float* __restrict__ out) {
    constexpr int STAGE  = 2 * LLG + 2 * LG;         // 8648 staged floats
    constexpr int STAGE4 = STAGE / 4;                // 2162 x b128
    __shared__ __attribute__((aligned(16))) float smem[STAGE + L + 2 * LG];

    const int tid   = threadIdx.x;
    const int p     = blockIdx.x / NCHUNK;
    const int chunk = blockIdx.x - p * NCHUNK;

    float* s_tpm = smem;                  // [c][q][g]  4232
    float* s_e2  = smem + LLG;            // [c][q][g]  4232
    float* s_tcm = smem + 2 * LLG;        // [c][g]       92 (slice at p)
    float* s_etc = smem + 2 * LLG + LG;   // [c][g]       92 (slice at p)
    float* s_csc = smem + STAGE;          // cs_scale[c]  46
    float* s_cmu = smem + STAGE + L;      // cs_mu[c][g]  92
    float* s_e1  = smem + STAGE + L + LG; // e1[c][g]     92

    // float2 views of the G=2-paired tables (all 8B aligned)
    const float2* s_tpm2 = (const float2*)s_tpm;   // index (c*L+q)
    const float2* s_e2v  = (const float2*)s_e2;    // index (c*L+q)
    const float2* s_tcm2 = (const float2*)s_tcm;   // index c
    const float2* s_etc2 = (const float2*)s_etc;   // index c
    const float2* s_cmuv = (const float2*)s_cmu;   // index c
    const float2* s_e1v  = (const float2*)s_e1;    // index c
    const float2* w_smv  = (const float2*)(wsf + WS_SM);  // index (bt*L+c)
    const float2* w_esv  = (const float2*)(wsf + WS_ES);  // index (bt*L+c)

    // ---- async-stage transition tables into LDS (b128 granules) ----
    // boundaries: 2*LLG=8464 and LG=92 are multiples of 4; slice strides
    // (p*LG*4 = 368B) are multiples of 16 -> every granule is 16B aligned.
    for (int i4 = tid; i4 < STAGE4; i4 += NTHR) {
        const int i = i4 * 4;
        const float* src;
        if (i < 2 * LLG)           src = wsf + i;                             // tpm,e2
        else if (i < 2 * LLG + LG) src = wsf + WS_TCM + p * LG + (i - 2 * LLG);
        else                       src = wsf + WS_ETC + p * LG + (i - 2 * LLG - LG);
        copy_g2lds_v4(src, smem + i);
    }
    async_wait0();
    __syncthreads();

    constexpr float LOG2PI = 1.8378770664093453f;    // ln(2*pi)

    for (int tt = 0; tt < TCHUNK; ++tt) {
        const int bt = chunk * TCHUNK + tt;
        const int t  = bt % T;
        const float m = mask[bt];

        // ---- child-state product for this parent p, all (c,g) ----
        if (tid < L) {
            const int c = tid;
            const float2 sm2 = w_smv[bt * L + c];    // tanh(s_mu), g=0/1
            const float2 v12 = w_esv[bt * L + c];    // exp(2*s_var)
            const float2 tc2 = s_tcm2[c];
            const float2 v22 = s_etc2[c];            // exp(2*t_c_var)
            float csum = 0.0f;
            float2 cmu, e1;
#pragma unroll
            for (int g = 0; g < G; ++g) {
                const float sm  = (g == 0) ? sm2.x : sm2.y;
                const float v1  = (g == 0) ? v12.x : v12.y;
                const float tcm = (g == 0) ? tc2.x : tc2.y;
                const float v2  = (g == 0) ? v22.x : v22.y;
                const float vadd = v1 + v2;
                const float rva  = fast_rcp(vadd);
                const float lva  = fast_log(vadd);
                const float d    = sm - tcm;
                csum += -0.5f * (LOG2PI + lva + d * d * rva);
                const float mu = (sm * v2 + tcm * v1) * rva;
                const float ee = v1 * v2 * rva;      // == exp(2*cs_var)
                if (g == 0) { cmu.x = mu; e1.x = ee; } else { cmu.y = mu; e1.y = ee; }
            }
            ((float2*)s_cmu)[c] = cmu;
            ((float2*)s_e1)[c]  = e1;
            s_csc[c] = csum;
        }
        __syncthreads();

        float* obase = out + (size_t)bt * LLL + (size_t)p * LL;  // 16B aligned

        if (t != T - 1) {
            // 2116 outputs = 529 exact float4 granules; oi = c*L+q indexes both
            // the output tile and (as float2) the [c][q][g] tables.
            for (int v4 = tid; v4 < LL / 4; v4 += NTHR) {
                const int oi0 = v4 * 4;
                vfloat4 r;
#pragma unroll
                for (int e = 0; e < 4; ++e) {
                    const int oi = oi0 + e;
                    const int c  = oi / L;
                    const float2 e1c = s_e1v[c];
                    const float2 muc = s_cmuv[c];
                    const float2 tp  = s_tpm2[oi];
                    const float2 ev  = s_e2v[oi];
                    const float V0 = e1c.x + ev.x;
                    const float V1 = e1c.y + ev.y;
                    const float d0 = muc.x - tp.x;
                    const float d1 = muc.y - tp.y;
                    float acc = 2.0f * LOG2PI
                              + fast_log(V0) + d0 * d0 * fast_rcp(V0)
                              + fast_log(V1) + d1 * d1 * fast_rcp(V1);
                    r[e] = (s_csc[c] - 0.5f * acc) * m;
                }
                __builtin_nontemporal_store(r, (vfloat4*)(obase + oi0));
            }
        } else {
            // last timestep: cs_scale broadcast over q
            for (int v4 = tid; v4 < LL / 4; v4 += NTHR) {
                const int oi0 = v4 * 4;
                vfloat4 r;
                r[0] = s_csc[(oi0 + 0) / L] * m;
                r[1] = s_csc[(oi0 + 1) / L] * m;
                r[2] = s_csc[(oi0 + 2) / L] * m;
                r[3] = s_csc[(oi0 + 3) / L] * m;
                __builtin_nontemporal_store(r, (vfloat4*)(obase + oi0));
            }
        }
        __syncthreads();   // s_csc/s_cmu/s_e1 rewritten next iteration
    }
}

// ---------------------------------------------------------------------------
extern "C" void kernel_launch(void* const* d_in, const int* in_sizes, int n_in,
                              void* d_out, int out_size, void* d_ws, size_t ws_size,
                              hipStream_t stream) {
    const int*   input = (const int*)d_in[0];
    const float* mask  = (const float*)d_in[1];
    const float* smt   = (const float*)d_in[2];
    const float* svt   = (const float*)d_in[3];
    const float* tpmu  = (const float*)d_in[4];
    const float* tpvar = (const float*)d_in[5];
    const float* tcmu  = (const float*)d_in[6];
    float* out = (float*)d_out;
    float* wsf = (float*)d_ws;   // needs 256128 bytes

    (void)in_sizes; (void)n_in; (void)out_size; (void)ws_size;

    lveg_prep_trans<<<(LLG + 255) / 256, 256, 0, stream>>>(tpmu, tpvar, tcmu, wsf);
    lveg_prep_emb<<<(BT * LG + 255) / 256, 256, 0, stream>>>(input, smt, svt, wsf);
    lveg_main<<<L * NCHUNK, NTHR, 0, stream>>>(wsf, mask, out);
}